// CapsuleLayer_54966991454535
// MI455X (gfx1250) — compile-verified
//
#include <hip/hip_runtime.h>
#include <math.h>

typedef __attribute__((ext_vector_type(2))) float v2f;
typedef __attribute__((ext_vector_type(8))) float v8f;

#define N_IN   91392     // 17*21*16*16
#define NPW    64        // capsule indices processed per wave
#define NBLK   (N_IN / NPW)   // 1428 single-wave workgroups
#define NSLOT  64        // partial-sum slots to spread atomic contention
#define SLOTF  (NSLOT * 512)

// Sum over each 16-lane half (xor masks <= 8 never cross the half boundary).
// After this every lane holds its half's total.
__device__ __forceinline__ float halfsum16(float t) {
    t += __shfl_xor(t, 1, 32);
    t += __shfl_xor(t, 2, 32);
    t += __shfl_xor(t, 4, 32);
    t += __shfl_xor(t, 8, 32);
    return t;
}

// One fused routing pass: recompute u_hat via WMMA f32 16x16x4 and accumulate
// s[b,j,o] = sum_n c[b,j,n] * u_hat[b,j,n,o].
// USE_V = false : iteration 0, c == 0.5 (applied once at the end) — the s
//                 accumulator IS the WMMA C/D operand, no copy-out per n.
// USE_V = true  : logits = vvec . u_hat (vvec = v0 for it1, v0+v1 for it2),
//                 c = 2-way softmax over j, s += c * u_hat.
template <bool USE_V>
__global__ __launch_bounds__(32)
void caps_pass(const float* __restrict__ x, const float* __restrict__ W,
               const float* __restrict__ vvec, float* __restrict__ slot)
{
    const int lane = threadIdx.x & 31;
    const int half = lane >> 4;      // selects K-half / batch-half
    const int lcol = lane & 15;      // A: batch row m ; B/C/D: column o
    const int n0   = blockIdx.x * NPW;

    // Preload v in the C/D accumulator layout: row r, lane l -> v[r+8*half, j, lcol]
    float v0r[8], v1r[8];
    if (USE_V) {
#pragma unroll
        for (int r = 0; r < 8; ++r) {
            int b = r + 8 * half;
            v0r[r] = vvec[(b * 2 + 0) * 16 + lcol];
            v1r[r] = vvec[(b * 2 + 1) * 16 + lcol];
        }
    }

    v8f s0 = {};   // s accumulator, j = 0 (C/D layout)
    v8f s1 = {};   // s accumulator, j = 1

    for (int k = 0; k < NPW; ++k) {
        const int n = n0 + k;

        // ---- A fragment (x): 16x4 f32, lane half picks K in {0,1}/{2,3} ----
        const float* xp = x + (lcol * N_IN + n) * 8 + 2 * half;
        v2f a1 = *(const v2f*)xp;         // K = klo, klo+1
        v2f a2 = *(const v2f*)(xp + 4);   // K = 4+klo, 5+klo

        // ---- B fragments (W): 4x16 f32, row k is a contiguous 64B line ----
        const float* wp0 = W + n * 128 + 2 * half * 16 + lcol;          // j = 0
        const float* wp1 = wp0 + N_IN * 128;                            // j = 1
        v2f b1, b2, b3, b4;
        b1.x = wp0[0];  b1.y = wp0[16];   // rows klo, klo+1
        b2.x = wp0[64]; b2.y = wp0[80];   // rows 4+klo, 5+klo
        b3.x = wp1[0];  b3.y = wp1[16];
        b4.x = wp1[64]; b4.y = wp1[80];

        if (!USE_V) {
            // Fold the n-sum directly into the WMMA accumulator: no per-n
            // copy-out / vector add needed.
            s0 = __builtin_amdgcn_wmma_f32_16x16x4_f32(false, a1, false, b1,
                                                       (short)0, s0, false, false);
            s0 = __builtin_amdgcn_wmma_f32_16x16x4_f32(false, a2, false, b2,
                                                       (short)0, s0, false, false);
            s1 = __builtin_amdgcn_wmma_f32_16x16x4_f32(false, a1, false, b3,
                                                       (short)0, s1, false, false);
            s1 = __builtin_amdgcn_wmma_f32_16x16x4_f32(false, a2, false, b4,
                                                       (short)0, s1, false, false);
        } else {
            // ---- u_hat tiles: D[b,o] per j, K=8 as two chained K=4 WMMAs ----
            v8f d0 = {};
            d0 = __builtin_amdgcn_wmma_f32_16x16x4_f32(false, a1, false, b1,
                                                       (short)0, d0, false, false);
            d0 = __builtin_amdgcn_wmma_f32_16x16x4_f32(false, a2, false, b2,
                                                       (short)0, d0, false, false);
            v8f d1 = {};
            d1 = __builtin_amdgcn_wmma_f32_16x16x4_f32(false, a1, false, b3,
                                                       (short)0, d1, false, false);
            d1 = __builtin_amdgcn_wmma_f32_16x16x4_f32(false, a2, false, b4,
                                                       (short)0, d1, false, false);
#pragma unroll
            for (int r = 0; r < 8; ++r) {
                // routing logits for (b = r+8*half, j): dot over o within half
                float t0 = halfsum16(v0r[r] * d0[r]);
                float t1 = halfsum16(v1r[r] * d1[r]);
                // 2-way softmax over j, per lane (already broadcast correctly)
                float c0 = 1.0f / (1.0f + __expf(t1 - t0));
                s0[r] = fmaf(c0,        d0[r], s0[r]);
                s1[r] = fmaf(1.0f - c0, d1[r], s1[r]);
            }
        }
    }

    // Wave covers all 512 distinct (b,j,o): 16 atomic adds, 32 lanes each.
    const float scale = USE_V ? 1.0f : 0.5f;   // iteration 0: c == 0.5
    float* sl = slot + (blockIdx.x & (NSLOT - 1)) * 512;
#pragma unroll
    for (int r = 0; r < 8; ++r) {
        int b = r + 8 * half;
        atomicAdd(&sl[(b * 2 + 0) * 16 + lcol], scale * s0[r]);
        atomicAdd(&sl[(b * 2 + 1) * 16 + lcol], scale * s1[r]);
    }
}

// Reduce slot partials, apply squash; optionally accumulate previous v so the
// next pass can use (v0 + v1) as its dot vector.
__global__ __launch_bounds__(512)
void caps_squash(const float* __restrict__ slot, const float* __restrict__ vadd,
                 float* __restrict__ vout)
{
    __shared__ float sh[512];
    const int t = threadIdx.x;             // t = ((b*2+j)*16 + o)
    float s = 0.0f;
    for (int k = 0; k < NSLOT; ++k) s += slot[k * 512 + t];
    sh[t] = s;
    __syncthreads();
    const int g = t >> 4;                  // (b,j) group
    float sn = 0.0f;
#pragma unroll
    for (int o = 0; o < 16; ++o) { float u = sh[(g << 4) + o]; sn = fmaf(u, u, sn); }
    float v = s * (sn / ((1.0f + sn) * sqrtf(sn + 1e-7f)));
    if (vadd) v += vadd[t];
    vout[t] = v;
}

__global__ void caps_zero(float* __restrict__ p, int n)
{
    int i = blockIdx.x * blockDim.x + threadIdx.x;
    if (i < n) p[i] = 0.0f;
}

extern "C" void kernel_launch(void* const* d_in, const int* in_sizes, int n_in,
                              void* d_out, int out_size, void* d_ws, size_t ws_size,
                              hipStream_t stream)
{
    const float* x = (const float*)d_in[0];   // [16, N_IN, 8]
    const float* W = (const float*)d_in[1];   // [2, N_IN, 8, 16]
    float* ws    = (float*)d_ws;
    float* slot0 = ws;
    float* slot1 = ws + SLOTF;
    float* slot2 = ws + 2 * SLOTF;
    float* v0    = ws + 3 * SLOTF;            // v from iteration 0
    float* vsum  = v0 + 512;                  // v0 + v1, dot vector for it 2

    const int zn = 3 * SLOTF;                 // slot buffers must start at zero
    caps_zero<<<(zn + 255) / 256, 256, 0, stream>>>(ws, zn);

    // it 0: c = 0.5 -> s0 -> v0
    caps_pass<false><<<NBLK, 32, 0, stream>>>(x, W, nullptr, slot0);
    caps_squash<<<1, 512, 0, stream>>>(slot0, nullptr, v0);
    // it 1: logits = v0 . u_hat -> s1 -> v1 ; stash vsum = v0 + v1
    caps_pass<true><<<NBLK, 32, 0, stream>>>(x, W, v0, slot1);
    caps_squash<<<1, 512, 0, stream>>>(slot1, v0, vsum);
    // it 2: logits = (v0+v1) . u_hat -> s2 -> output v
    caps_pass<true><<<NBLK, 32, 0, stream>>>(x, W, vsum, slot2);
    caps_squash<<<1, 512, 0, stream>>>(slot2, nullptr, (float*)d_out);
}